// HGTModel_12498354831933
// MI455X (gfx1250) — compile-verified
//
#include <hip/hip_runtime.h>
#include <math.h>

// ============================================================================
// HGT forward for MI455X (gfx1250, wave32).
// - Dense projections + per-head relation transforms: fp32 WMMA GEMM
//   (V_WMMA_F32_16X16X4_F32). One wave owns a 16 x (NT*16) output strip:
//   A-fragment reused across NT column tiles, B loads use compile-time LDB
//   (immediate offsets after unroll), 8 v8f accumulators for the 128-wide case.
// - Edge stage: logits + segment-softmax (uint-encoded atomicMax, atomicAdd)
//   + fused alpha*vr scatter with global fp32 atomics (L2-resident tables).
// - LayerNorm/skip: one wave per 128-elem row, shfl_xor reductions.
// ============================================================================

typedef float v2f __attribute__((ext_vector_type(2)));
typedef float v8f __attribute__((ext_vector_type(8)));

#define HID   128
#define HEADS 4
#define DH    32

// ---------------------------------------------------------------------------
// fp32 WMMA GEMM strip kernel: C[M, NT*16] = A[M,K] @ B[K, NT*16] (+bias),
// batched over blockIdx.z with element strides. LDB is compile-time so the
// unrolled K-loop uses immediate load offsets.
//
// Fragment layouts (ISA 7.12.2):
//  A (16x4): lanes 0-15 -> M=lane, K={kk,kk+1}; lanes 16-31 -> K={kk+2,kk+3}
//  B (4x16): VGPR0: K=kk (lanes0-15) / K=kk+2 (lanes16-31); VGPR1: +1
//  C (16x16): VGPR i -> M=tileM+i+8*(lane>>4), N=col0+(lane&15)
// ---------------------------------------------------------------------------
template <int NT, int LDB>
__global__ __launch_bounds__(128) void wmma_gemm_strip(
    const float* __restrict__ A, int lda, int strideA,
    const float* __restrict__ B, int strideB,
    float*       __restrict__ C, int ldc, int strideC,
    const float* __restrict__ bias,
    int M, int K)
{
  const int lane  = threadIdx.x & 31;
  const int wave  = threadIdx.x >> 5;
  const int tileM = (blockIdx.y * (blockDim.x >> 5) + wave) * 16;
  if (tileM >= M) return;            // wave-uniform: EXEC stays all-ones

  const int z = blockIdx.z;
  A += (size_t)z * (size_t)strideA;
  B += (size_t)z * (size_t)strideB;
  C += (size_t)z * (size_t)strideC;

  const int half = lane >> 4;        // 0 -> K pair {0,1}, 1 -> K pair {2,3}
  const int l16  = lane & 15;

  const float* aRow  = A + (size_t)(tileM + l16) * (size_t)lda + 2 * half;
  const float* bBase = B + (size_t)(2 * half) * LDB + l16;

  v8f acc[NT];
#pragma unroll
  for (int t = 0; t < NT; ++t) acc[t] = (v8f){};

#pragma unroll 4
  for (int kk = 0; kk < K; kk += 4) {
    v2f a;
    a.x = aRow[kk + 0];
    a.y = aRow[kk + 1];
#pragma unroll
    for (int t = 0; t < NT; ++t) {
      v2f b;
      b.x = bBase[(kk + 0) * LDB + t * 16];
      b.y = bBase[(kk + 1) * LDB + t * 16];
      acc[t] = __builtin_amdgcn_wmma_f32_16x16x4_f32(
          /*neg_a=*/false, a, /*neg_b=*/false, b,
          /*c_mod=*/(short)0, acc[t], /*reuse_a=*/false, /*reuse_b=*/false);
    }
  }

#pragma unroll
  for (int t = 0; t < NT; ++t) {
    const int   col = t * 16 + l16;
    const float bb  = bias ? bias[col] : 0.0f;
#pragma unroll
    for (int i = 0; i < 8; ++i) {
      const int row = tileM + i + 8 * half;
      C[(size_t)row * (size_t)ldc + col] = acc[t][i] + bb;
    }
  }
}

// ---------------------------------------------------------------------------
// Edge pass 1: logits[e,h] = dot(q[dst,h,:], kr[src,h,:]) * prel[h] * 1/sqrt(d)
// and segment-max over dst via monotone uint encoding + atomicMax.
// ---------------------------------------------------------------------------
__device__ __forceinline__ unsigned enc_f32(float f) {
  unsigned b = __float_as_uint(f);
  return (b & 0x80000000u) ? ~b : (b | 0x80000000u);
}
__device__ __forceinline__ float dec_f32(unsigned e) {
  return (e & 0x80000000u) ? __uint_as_float(e ^ 0x80000000u)
                           : __uint_as_float(~e);
}

__global__ __launch_bounds__(256) void edge_logits_kernel(
    const int* __restrict__ ei, int E,
    const float* __restrict__ q,   // full node table [NODES,128]
    const float* __restrict__ kr,  // per-relation src table [n_src,128]
    int dstOff,
    const float* __restrict__ prel,
    float* __restrict__ logits,          // [E,HEADS]
    unsigned* __restrict__ mEnc,         // [n_dst,HEADS], init 0
    float scale)
{
  long t = (long)blockIdx.x * blockDim.x + threadIdx.x;
  if (t >= (long)E * HEADS) return;
  int e = (int)(t >> 2);
  int h = (int)(t & 3);
  int src = ei[e];
  int dst = ei[E + e];

  const float4* qp = (const float4*)(q  + ((size_t)(dstOff + dst)) * HID + h * DH);
  const float4* kp = (const float4*)(kr + ((size_t)src) * HID + h * DH);
  float acc = 0.0f;
#pragma unroll
  for (int i = 0; i < DH / 4; ++i) {
    float4 a = qp[i], b = kp[i];
    acc += a.x * b.x + a.y * b.y + a.z * b.z + a.w * b.w;
  }
  acc *= prel[h] * scale;
  logits[t] = acc;
  atomicMax(&mEnc[(size_t)dst * HEADS + h], enc_f32(acc));
}

// ---------------------------------------------------------------------------
// Edge pass 2: e = exp(logit - m[dst]); segment-sum via atomicAdd.
// ---------------------------------------------------------------------------
__global__ __launch_bounds__(256) void edge_exp_kernel(
    const int* __restrict__ ei, int E,
    float* __restrict__ logits,
    const unsigned* __restrict__ mEnc,
    float* __restrict__ ssum)            // [n_dst,HEADS], init 0
{
  long t = (long)blockIdx.x * blockDim.x + threadIdx.x;
  if (t >= (long)E * HEADS) return;
  int e = (int)(t >> 2);
  int h = (int)(t & 3);
  int dst = ei[E + e];
  float m = dec_f32(mEnc[(size_t)dst * HEADS + h]);
  if (!isfinite(m)) m = 0.0f;   // matches reference where(isfinite(m), m, 0)
  float ev = expf(logits[t] - m);
  logits[t] = ev;
  atomicAdd(&ssum[(size_t)dst * HEADS + h], ev);
}

// ---------------------------------------------------------------------------
// Edge pass 3: agg[dst, c] += vr[src, c] * alpha[e, c/32]
// ---------------------------------------------------------------------------
__global__ __launch_bounds__(256) void edge_scatter_kernel(
    const int* __restrict__ ei, int E,
    const float* __restrict__ logits,    // now holds exp values
    const float* __restrict__ ssum,
    const float* __restrict__ vr,        // [n_src,128]
    float* __restrict__ agg,             // [NODES,128]
    int dstOff)
{
  long t = (long)blockIdx.x * blockDim.x + threadIdx.x;
  if (t >= (long)E * HID) return;
  int e = (int)(t >> 7);
  int c = (int)(t & 127);
  int h = c >> 5;
  int src = ei[e];
  int dst = ei[E + e];
  float alpha = logits[(size_t)e * HEADS + h] /
                (ssum[(size_t)dst * HEADS + h] + 1e-16f);
  atomicAdd(&agg[((size_t)(dstOff + dst)) * HID + c],
            vr[(size_t)src * HID + c] * alpha);
}

// ---------------------------------------------------------------------------
// Elementwise exact GELU: 0.5*x*(1+erf(x/sqrt(2)))
// ---------------------------------------------------------------------------
__global__ __launch_bounds__(256) void gelu_kernel(
    const float* __restrict__ in, float* __restrict__ out, long n)
{
  long t = (long)blockIdx.x * blockDim.x + threadIdx.x;
  if (t >= n) return;
  float x = in[t];
  out[t] = 0.5f * x * (1.0f + erff(x * 0.70710678118654752440f));
}

// ---------------------------------------------------------------------------
// Gated skip + residual + LayerNorm. One wave per row (128 cols, 4/lane).
//   o2 = sk*o + (1-sk)*x ;  y = LN(o2 + x)*g + b
// ---------------------------------------------------------------------------
__global__ __launch_bounds__(256) void skip_ln_kernel(
    const float* __restrict__ o,    // [n,128] local rows
    const float* __restrict__ x,    // [n,128]
    const float* __restrict__ skip, // scalar
    const float* __restrict__ g,
    const float* __restrict__ b,
    float* __restrict__ y,          // [n,128]
    int n)
{
  int row  = blockIdx.x * (blockDim.x >> 5) + (threadIdx.x >> 5);
  if (row >= n) return;
  int lane = threadIdx.x & 31;
  float sk = 1.0f / (1.0f + expf(-skip[0]));

  float v[4];
  float sum = 0.0f;
#pragma unroll
  for (int i = 0; i < 4; ++i) {
    int c = lane * 4 + i;
    float oo = o[(size_t)row * HID + c];
    float xx = x[(size_t)row * HID + c];
    float t  = sk * oo + (1.0f - sk) * xx + xx;
    v[i] = t;
    sum += t;
  }
#pragma unroll
  for (int off = 16; off > 0; off >>= 1) sum += __shfl_xor(sum, off, 32);
  float mu = sum * (1.0f / HID);

  float vs = 0.0f;
#pragma unroll
  for (int i = 0; i < 4; ++i) { float d = v[i] - mu; vs += d * d; }
#pragma unroll
  for (int off = 16; off > 0; off >>= 1) vs += __shfl_xor(vs, off, 32);
  float rstd = rsqrtf(vs * (1.0f / HID) + 1e-5f);

#pragma unroll
  for (int i = 0; i < 4; ++i) {
    int c = lane * 4 + i;
    y[(size_t)row * HID + c] = (v[i] - mu) * rstd * g[c] + b[c];
  }
}

// ============================================================================
// Host orchestration
// ============================================================================
extern "C" void kernel_launch(void* const* d_in, const int* in_sizes, int n_in,
                              void* d_out, int out_size, void* d_ws, size_t ws_size,
                              hipStream_t stream)
{
  (void)n_in; (void)out_size; (void)ws_size;

  // ---- graph constants (match reference) ----
  const int NCNT[3] = {50000, 20000, 10000};   // drug, side_effect, biomarker
  const int NOFF[3] = {0, 50000, 70000};
  const int FIN[3]  = {256, 128, 64};
  const int NODES   = 80000;
  const int ESRC[4] = {0, 1, 0, 2};            // causes, rev_causes, affects, rev_affects
  const int EDST[4] = {1, 0, 2, 0};
  const float INV_SQRT_D = 0.17677669529663688f;

  // ---- param index map (insertion order of _make_params) ----
  // 0..2: x_*, 3..6: ei_*, 7..12: in_W/in_b per node type,
  // per layer (45): 3 node types x [kW,kb,qW,qb,vW,vb,aW,ab,skip,ln_g,ln_b](11)
  //               + 4 relations x [arel, mrel, prel](3)
  auto P = [&](int i) -> const float* { return (const float*)d_in[i]; };
  auto layerBase = [&](int l) { return 13 + 45 * l; };
  auto ntBase    = [&](int l, int t) { return layerBase(l) + 11 * t; };
  auto relBase   = [&](int l, int j) { return layerBase(l) + 33 + 3 * j; };

  // ---- workspace layout (floats) ----
  float* X0  = (float*)d_ws;                       // [NODES,128]
  float* X1  = X0  + (size_t)NODES * HID;
  float* Kb  = X1  + (size_t)NODES * HID;
  float* Qb  = Kb  + (size_t)NODES * HID;
  float* Vb  = Qb  + (size_t)NODES * HID;
  float* AGG = Vb  + (size_t)NODES * HID;
  float* TMP = AGG + (size_t)NODES * HID;          // gelu output
  float* KR  = TMP + (size_t)NODES * HID;          // [<=50000,128], also reused as 'o'
  float* VR  = KR  + (size_t)50000 * HID;
  float* LOG = VR  + (size_t)50000 * HID;          // [<=1M, HEADS]
  unsigned* MENC = (unsigned*)(LOG + (size_t)1000000 * HEADS); // [<=50000,HEADS]
  float*    SSUM = (float*)(MENC + (size_t)50000 * HEADS);

  // Dense GEMM: N = 128 (8 column tiles per wave), LDB = 128.
  auto gemmDense = [&](const float* A, int lda,
                       const float* B, float* C,
                       const float* bias, int M, int K) {
    dim3 blk(128);
    dim3 grd(1, (M / 16 + 3) / 4, 1);
    wmma_gemm_strip<8, HID><<<grd, blk, 0, stream>>>(
        A, lda, 0, B, 0, C, HID, 0, bias, M, K);
  };
  // Relation transform: per-head [M,32] @ [32,32], batched z over 4 heads.
  auto gemmRel = [&](const float* A, const float* B, float* C, int M) {
    dim3 blk(128);
    dim3 grd(1, (M / 16 + 3) / 4, HEADS);
    wmma_gemm_strip<2, DH><<<grd, blk, 0, stream>>>(
        A, HID, DH, B, DH * DH, C, HID, DH, nullptr, M, DH);
  };

  // ---- input projections: X0 = x_nt @ in_W_nt + in_b_nt ----
  for (int t = 0; t < 3; ++t) {
    gemmDense((const float*)d_in[t], FIN[t], P(7 + 2 * t),
              X0 + (size_t)NOFF[t] * HID, P(8 + 2 * t), NCNT[t], FIN[t]);
  }

  // ---- layers ----
  for (int l = 0; l < 2; ++l) {
    float* xc = (l == 0) ? X0 : X1;
    float* xn = (l == 0) ? X1 : (float*)d_out;

    // k/q/v projections per node type
    for (int t = 0; t < 3; ++t) {
      const int nb = ntBase(l, t);
      const size_t off = (size_t)NOFF[t] * HID;
      gemmDense(xc + off, HID, P(nb + 0), Kb + off, P(nb + 1), NCNT[t], HID);
      gemmDense(xc + off, HID, P(nb + 2), Qb + off, P(nb + 3), NCNT[t], HID);
      gemmDense(xc + off, HID, P(nb + 4), Vb + off, P(nb + 5), NCNT[t], HID);
    }

    hipMemsetAsync(AGG, 0, (size_t)NODES * HID * sizeof(float), stream);

    // relations
    for (int j = 0; j < 4; ++j) {
      const int st = ESRC[j], dt = EDST[j];
      const int E  = in_sizes[3 + j] / 2;
      const int* ei = (const int*)d_in[3 + j];
      const int rb = relBase(l, j);
      const int nSrc = NCNT[st], nDst = NCNT[dt];

      // kr = k[st] x arel, vr = v[st] x mrel  (batched over 4 heads)
      gemmRel(Kb + (size_t)NOFF[st] * HID, P(rb + 0), KR, nSrc);
      gemmRel(Vb + (size_t)NOFF[st] * HID, P(rb + 1), VR, nSrc);

      hipMemsetAsync(MENC, 0, (size_t)nDst * HEADS * sizeof(unsigned), stream);
      hipMemsetAsync(SSUM, 0, (size_t)nDst * HEADS * sizeof(float), stream);

      long nEH = (long)E * HEADS;
      edge_logits_kernel<<<(unsigned)((nEH + 255) / 256), 256, 0, stream>>>(
          ei, E, Qb, KR, NOFF[dt], P(rb + 2), LOG, MENC, INV_SQRT_D);
      edge_exp_kernel<<<(unsigned)((nEH + 255) / 256), 256, 0, stream>>>(
          ei, E, LOG, MENC, SSUM);
      long nEC = (long)E * HID;
      edge_scatter_kernel<<<(unsigned)((nEC + 255) / 256), 256, 0, stream>>>(
          ei, E, LOG, SSUM, VR, AGG, NOFF[dt]);
    }

    // output transform + gated skip + layernorm per node type
    for (int t = 0; t < 3; ++t) {
      const int nb = ntBase(l, t);
      const size_t off = (size_t)NOFF[t] * HID;
      const long n128 = (long)NCNT[t] * HID;

      gelu_kernel<<<(unsigned)((n128 + 255) / 256), 256, 0, stream>>>(
          AGG + off, TMP + off, n128);
      // o = gelu(agg) @ a_W + a_b   (into KR scratch, local rows)
      gemmDense(TMP + off, HID, P(nb + 6), KR, P(nb + 7), NCNT[t], HID);
      skip_ln_kernel<<<(NCNT[t] + 7) / 8, 256, 0, stream>>>(
          KR, xc + off, P(nb + 8), P(nb + 9), P(nb + 10), xn + off, NCNT[t]);
    }
  }
}